// ResidualGatedGCNLayer_549755814509
// MI455X (gfx1250) — compile-verified
//
#include <hip/hip_runtime.h>

typedef __attribute__((ext_vector_type(2))) float v2f;
typedef __attribute__((ext_vector_type(8))) float v8f;

#define DEVFN __device__ __forceinline__

constexpr int Bb  = 4, Nn = 96, Kk = 8, Hh = 128;
constexpr int BNK = Bb * Nn * Kk;            // 3072 node rows
constexpr int SLAB_ROWS = Nn * Kk;           // 768 edge rows per (b,i)
constexpr long RE = (long)Bb * Nn * Nn * Kk; // 294912 edge rows
constexpr float EPS = 1e-5f;

// ---------------- WMMA f32 16x16x4 helpers ----------------

DEVFN v8f wmma4(v2f a, v2f b, v8f c) {
  // 8 args: (neg_a, A, neg_b, B, c_mod, C, reuse_a, reuse_b)
  return __builtin_amdgcn_wmma_f32_16x16x4_f32(false, a, false, b, (short)0, c,
                                               false, false);
}

// Weight staged in LDS in per-lane B-operand layout:
//   WB[((ks/4)*8 + nt)*64 + lane*2 + c] = W[n][k]   (out = A @ W^T)
//   where n = nt*16 + (lane&15), k = (ks/4)*4 + 2*(lane>>4) + c
// => each lane's B operand for (ks, nt) is one aligned 8-byte LDS load.
DEVFN void stage_wb(const float* __restrict__ W, float* WB, int tid, int nthr) {
  for (int idx = tid; idx < Hh * Hh; idx += nthr) {
    const int p    = idx & 63;       // position within 64-float group
    const int g    = idx >> 6;       // group = ks4*8 + nt
    const int nt   = g & 7;
    const int ks4  = g >> 3;
    const int lane = p >> 1;
    const int c    = p & 1;
    const int n    = nt * 16 + (lane & 15);
    const int k    = ks4 * 4 + ((lane >> 4) << 1) + c;
    WB[idx] = W[n * Hh + k];
  }
}

// Compute a 16-row x 128-col strip:  acc += A(16x128) @ W^T
// A: row-major, ld = Hh.  WB: LDS, swizzled layout above.
// Two-deep software pipeline on the B operands: while WMMAs consume bq0,
// the DS loads for bq1 (next k-step) are in flight. bq0 stays live across
// the bq1 loads, so the allocator cannot recycle the registers, and each
// run of 8 WMMAs issues behind a single coarse s_wait_dscnt.
DEVFN void gemm_strip(const float* __restrict__ A, const float* __restrict__ WB,
                      v8f acc[8], int lane) {
  const int mrow = lane & 15;
  const int kb   = (lane >> 4) << 1;   // lane half selects K pair {0,1}/{2,3}
  const int l2   = lane << 1;
  const float* arow = A + mrow * Hh + kb;
  const float* wl   = WB + l2;

  v2f bq0[8], bq1[8];
#pragma unroll
  for (int nt = 0; nt < 8; ++nt)                       // prologue: k-step 0
    bq0[nt] = *(const v2f*)(wl + nt * 64);

  for (int ks = 0; ks < Hh; ks += 8) {                 // 16 iters, 2 k-steps each
    v2f a0 = *(const v2f*)(arow + ks);
    const float* w1 = wl + ((ks >> 2) + 1) * 512;
#pragma unroll
    for (int nt = 0; nt < 8; ++nt)                     // prefetch k-step s+1
      bq1[nt] = *(const v2f*)(w1 + nt * 64);
#pragma unroll
    for (int nt = 0; nt < 8; ++nt)                     // consume k-step s
      acc[nt] = wmma4(a0, bq0[nt], acc[nt]);

    v2f a1 = *(const v2f*)(arow + ks + 4);
    if (ks + 8 < Hh) {
      const float* w2 = wl + ((ks >> 2) + 2) * 512;
#pragma unroll
      for (int nt = 0; nt < 8; ++nt)                   // prefetch k-step s+2
        bq0[nt] = *(const v2f*)(w2 + nt * 64);
    }
#pragma unroll
    for (int nt = 0; nt < 8; ++nt)                     // consume k-step s+1
      acc[nt] = wmma4(a1, bq1[nt], acc[nt]);
  }
}

// ---------------- Kernel 1: node linears (Vx_e, Vx_n, Ux_n) ----------------
// grid: 3 matrices * 24 blocks; block = 256 thr (8 waves); wave -> one 16x128 strip
__global__ void node_lin_kernel(const float* __restrict__ x,
                                const float* __restrict__ Ve_w, const float* __restrict__ Ve_b,
                                const float* __restrict__ Vn_w, const float* __restrict__ Vn_b,
                                const float* __restrict__ Un_w, const float* __restrict__ Un_b,
                                float* __restrict__ Vxe, float* __restrict__ Vxn,
                                float* __restrict__ Uxn) {
  extern __shared__ float smem[];   // WB: Hh*Hh floats (64 KB)
  const int which = blockIdx.x / 24;
  const int blk   = blockIdx.x % 24;
  const float* W   = (which == 0) ? Ve_w : (which == 1) ? Vn_w : Un_w;
  const float* bia = (which == 0) ? Ve_b : (which == 1) ? Vn_b : Un_b;
  float* out       = (which == 0) ? Vxe  : (which == 1) ? Vxn  : Uxn;

  const int tid = threadIdx.x;
  stage_wb(W, smem, tid, blockDim.x);
  __syncthreads();

  const int wave = tid >> 5, lane = tid & 31;
  const int m0 = (blk * 8 + wave) * 16;      // 24*8*16 = 3072 rows

  v8f acc[8] = {};
  gemm_strip(x + (long)m0 * Hh, smem, acc, lane);

  const int half = lane >> 4, nlo = lane & 15;
#pragma unroll
  for (int nt = 0; nt < 8; ++nt) {
    const int n = nt * 16 + nlo;
    const float bv = bia[n];
#pragma unroll
    for (int r = 0; r < 8; ++r) {
      const int m = m0 + r + 8 * half;
      out[(long)m * Hh + n] = acc[nt][r] + bv;
    }
  }
}

// ---------------- Kernel 2/4: edge pass ----------------
// One block per (b,i) slab of 768 rows. PASS 0: BN stats + gated aggregation.
// PASS 1: recompute GEMM, normalize + ReLU + residual, write e_new.
template <int PASS>
__global__ void edge_pass_kernel(const float* __restrict__ e,
                                 const float* __restrict__ Ue_w,
                                 const float* __restrict__ Ue_b,
                                 const float* __restrict__ Vxe,
                                 const float* __restrict__ Vxn,
                                 float* __restrict__ x_acc,
                                 float* __restrict__ e_sum, float* __restrict__ e_sqs,
                                 const float* __restrict__ e_scale,
                                 const float* __restrict__ e_shift,
                                 float* __restrict__ e_out) {
  extern __shared__ float smem[];
  float* WB  = smem;                         // Hh*Hh (64 KB)
  float* xg  = smem + Hh * Hh;               // Kk*Hh  (PASS 0 gate accum)
  float* est = xg + Kk * Hh;                 // 2*Hh   (PASS 0 stats accum)

  const int bi = blockIdx.x;                 // 0 .. B*N-1
  const int b = bi / Nn, i = bi % Nn;
  const int tid = threadIdx.x;
  const int wave = tid >> 5, lane = tid & 31;

  stage_wb(Ue_w, WB, tid, blockDim.x);
  if (PASS == 0) {
    for (int idx = tid; idx < Kk * Hh; idx += blockDim.x) xg[idx] = 0.f;
    for (int idx = tid; idx < 2 * Hh;  idx += blockDim.x) est[idx] = 0.f;
  }
  __syncthreads();

  const float* eslab = e + (long)bi * SLAB_ROWS * Hh;
  const int half = lane >> 4, nlo = lane & 15;

  float s_sum[8] = {0.f, 0.f, 0.f, 0.f, 0.f, 0.f, 0.f, 0.f};
  float s_sq [8] = {0.f, 0.f, 0.f, 0.f, 0.f, 0.f, 0.f, 0.f};

  for (int t = wave; t < SLAB_ROWS / 16; t += 8) {   // 48 tiles / 8 waves
    const int m0 = t * 16;
    const int j = (m0 >> 3) + half;          // tile spans 2 j values; k == r
    v8f acc[8] = {};
    gemm_strip(eslab + (long)m0 * Hh, WB, acc, lane);

#pragma unroll
    for (int nt = 0; nt < 8; ++nt) {
      const int n = nt * 16 + nlo;
      const float bias = Ue_b[n];
      float sc = 0.f, sh = 0.f;
      if (PASS == 1) { sc = e_scale[n]; sh = e_shift[n]; }
#pragma unroll
      for (int r = 0; r < 8; ++r) {          // r == k for this tile shape
        const float vxi = Vxe[((b * Nn + i) * Kk + r) * Hh + n];
        const float vxj = Vxe[((b * Nn + j) * Kk + r) * Hh + n];
        const float tv = acc[nt][r] + bias + vxi + vxj;
        if (PASS == 0) {
          s_sum[nt] += tv;
          s_sq[nt]  += tv * tv;
          const float g = 1.f / (1.f + __expf(-tv));
          atomicAdd(&xg[r * Hh + n], g * Vxn[((b * Nn + j) * Kk + r) * Hh + n]);
        } else {
          float y = tv * sc + sh;
          y = (y > 0.f) ? y : 0.f;
          const long off = (long)bi * SLAB_ROWS * Hh + (long)(m0 + r + 8 * half) * Hh + n;
          e_out[off] = y + e[off];
        }
      }
    }
  }

  if (PASS == 0) {
    // fold per-lane stat partials into LDS, then one global atomic per channel
#pragma unroll
    for (int nt = 0; nt < 8; ++nt) {
      const int n = nt * 16 + nlo;
      atomicAdd(&est[n], s_sum[nt]);
      atomicAdd(&est[Hh + n], s_sq[nt]);
    }
    __syncthreads();
    for (int idx = tid; idx < Kk * Hh; idx += blockDim.x)
      x_acc[(long)bi * Kk * Hh + idx] = xg[idx];
    for (int idx = tid; idx < Hh; idx += blockDim.x) {
      atomicAdd(&e_sum[idx], est[idx]);
      atomicAdd(&e_sqs[idx], est[Hh + idx]);
    }
  }
}

// ---------------- small kernels ----------------

__global__ void init_stats_kernel(float* p) { p[threadIdx.x] = 0.f; }  // <<<1,512>>>

__global__ void bn_prep_kernel(const float* __restrict__ sum, const float* __restrict__ sqs,
                               float count, const float* __restrict__ g,
                               const float* __restrict__ bta,
                               float* __restrict__ scale, float* __restrict__ shift) {
  const int h = threadIdx.x;
  const float m = sum[h] / count;
  const float v = sqs[h] / count - m * m;
  const float sc = g[h] * rsqrtf(v + EPS);
  scale[h] = sc;
  shift[h] = bta[h] - m * sc;
}

__global__ void node_tmp_kernel(const float* __restrict__ Uxn, const float* __restrict__ x_acc,
                                float* __restrict__ x_tmp,
                                float* __restrict__ n_sum, float* __restrict__ n_sqs) {
  const int h = threadIdx.x;                 // 128 channels
  const int r0 = blockIdx.x * 24;            // 128 blocks * 24 rows = 3072
  float s = 0.f, q = 0.f;
  for (int r = r0; r < r0 + 24; ++r) {
    const int idx = r * Hh + h;
    const float v = Uxn[idx] + x_acc[idx];
    x_tmp[idx] = v;
    s += v;
    q += v * v;
  }
  atomicAdd(&n_sum[h], s);
  atomicAdd(&n_sqs[h], q);
}

__global__ void node_out_kernel(const float* __restrict__ x_tmp,
                                const float* __restrict__ sc, const float* __restrict__ sh,
                                const float* __restrict__ x, float* __restrict__ x_out) {
  const int idx = blockIdx.x * blockDim.x + threadIdx.x;   // BNK*Hh
  const int h = idx & (Hh - 1);
  float y = x_tmp[idx] * sc[h] + sh[h];
  y = (y > 0.f) ? y : 0.f;
  x_out[idx] = y + x[idx];
}

// ---------------- launcher ----------------

extern "C" void kernel_launch(void* const* d_in, const int* in_sizes, int n_in,
                              void* d_out, int out_size, void* d_ws, size_t ws_size,
                              hipStream_t stream) {
  (void)in_sizes; (void)n_in; (void)out_size; (void)ws_size;

  const float* x    = (const float*)d_in[0];
  const float* e    = (const float*)d_in[1];
  const float* Ue_w = (const float*)d_in[2];
  const float* Ue_b = (const float*)d_in[3];
  const float* Ve_w = (const float*)d_in[4];
  const float* Ve_b = (const float*)d_in[5];
  const float* Un_w = (const float*)d_in[6];
  const float* Un_b = (const float*)d_in[7];
  const float* Vn_w = (const float*)d_in[8];
  const float* Vn_b = (const float*)d_in[9];
  const float* bn_e_g = (const float*)d_in[10];
  const float* bn_e_b = (const float*)d_in[11];
  const float* bn_n_g = (const float*)d_in[12];
  const float* bn_n_b = (const float*)d_in[13];

  float* x_out = (float*)d_out;             // (B,N,K,H)
  float* e_out = x_out + (long)BNK * Hh;    // (B,N,N,K,H)

  // workspace partition (floats)
  float* ws     = (float*)d_ws;
  float* Vxe    = ws;                       // BNK*Hh
  float* Vxn    = Vxe   + (long)BNK * Hh;
  float* Uxn    = Vxn   + (long)BNK * Hh;
  float* x_acc  = Uxn   + (long)BNK * Hh;
  float* x_tmp  = x_acc + (long)BNK * Hh;
  float* e_sum  = x_tmp + (long)BNK * Hh;   // 4*Hh contiguous stats
  float* e_sqs  = e_sum + Hh;
  float* n_sum  = e_sqs + Hh;
  float* n_sqs  = n_sum + Hh;
  float* e_scale = n_sqs  + Hh;
  float* e_shift = e_scale + Hh;
  float* n_scale = e_shift + Hh;
  float* n_shift = n_scale + Hh;

  const int LIN_SMEM  = Hh * Hh * (int)sizeof(float);                        // 65,536 B
  const int EDGE_SMEM = (Hh * Hh + Kk * Hh + 2 * Hh) * (int)sizeof(float);   // 70,656 B

  (void)hipFuncSetAttribute(reinterpret_cast<const void*>(&node_lin_kernel),
                            hipFuncAttributeMaxDynamicSharedMemorySize, LIN_SMEM);
  (void)hipFuncSetAttribute(reinterpret_cast<const void*>(&edge_pass_kernel<0>),
                            hipFuncAttributeMaxDynamicSharedMemorySize, EDGE_SMEM);
  (void)hipFuncSetAttribute(reinterpret_cast<const void*>(&edge_pass_kernel<1>),
                            hipFuncAttributeMaxDynamicSharedMemorySize, EDGE_SMEM);

  // 1) zero BN stat accumulators (e_sum..n_sqs are contiguous, 512 floats)
  init_stats_kernel<<<1, 4 * Hh, 0, stream>>>(e_sum);

  // 2) node linears: Vx_e, Vx_n, Ux_n   (3 * 192 strips / 8 waves = 72 blocks)
  node_lin_kernel<<<72, 256, LIN_SMEM, stream>>>(x, Ve_w, Ve_b, Vn_w, Vn_b,
                                                 Un_w, Un_b, Vxe, Vxn, Uxn);

  // 3) edge pass 0: BN stats + gated aggregation -> x_acc
  edge_pass_kernel<0><<<Bb * Nn, 256, EDGE_SMEM, stream>>>(
      e, Ue_w, Ue_b, Vxe, Vxn, x_acc, e_sum, e_sqs, nullptr, nullptr, nullptr);

  // 4) edge BN scale/shift
  bn_prep_kernel<<<1, Hh, 0, stream>>>(e_sum, e_sqs, (float)RE,
                                       bn_e_g, bn_e_b, e_scale, e_shift);

  // 5) x_tmp = Ux_n + x_acc, node BN stats
  node_tmp_kernel<<<Hh, Hh, 0, stream>>>(Uxn, x_acc, x_tmp, n_sum, n_sqs);

  // 6) node BN scale/shift
  bn_prep_kernel<<<1, Hh, 0, stream>>>(n_sum, n_sqs, (float)BNK,
                                       bn_n_g, bn_n_b, n_scale, n_shift);

  // 7) x_new = relu(bn(x_tmp)) + x
  node_out_kernel<<<(BNK * Hh) / 256, 256, 0, stream>>>(x_tmp, n_scale, n_shift,
                                                        x, x_out);

  // 8) edge pass 1: recompute GEMM (e mostly L2-resident), normalize+relu+residual
  edge_pass_kernel<1><<<Bb * Nn, 256, EDGE_SMEM, stream>>>(
      e, Ue_w, Ue_b, Vxe, Vxn, nullptr, nullptr, nullptr, e_scale, e_shift, e_out);
}